// DeepClusteringLoss_24713241821608
// MI455X (gfx1250) — compile-verified
//
#include <hip/hip_runtime.h>

// Problem constants (from reference): B=8, F=256, T=512, D=40, S=4
#define BB 8
#define NN 131072          // F*T
#define DD 40
#define SS 4
#define WCOLS 48           // padded columns of W = [V | E | pad]
#define NCHUNK 64          // chunks per batch (512 blocks total -> 4096 waves)
#define ROWS_PER_BLOCK (NN / NCHUNK)            // 2048
#define NWAVES 8
#define ROWS_PER_WAVE (ROWS_PER_BLOCK / NWAVES) // 256
#define NSTEPS (ROWS_PER_WAVE / 4)              // 64 (K=4 per WMMA)
#define NTILE 6            // upper-triangular 16x16 tiles of 48x48 G
#define TILEVALS 256       // 8 VGPRs * 32 lanes
#define PARTIAL_FLOATS (NTILE * TILEVALS)       // 1536 floats per block

typedef float v2f __attribute__((ext_vector_type(2)));
typedef float v8f __attribute__((ext_vector_type(8)));

// D = A(16x4, f32) * B(4x16, f32) + C(16x16, f32)
__device__ __forceinline__ v8f wmma_f32(v2f a, v2f b, v8f c) {
    return __builtin_amdgcn_wmma_f32_16x16x4_f32(
        /*neg_a=*/false, a, /*neg_b=*/false, b,
        /*c_mod=*/(short)0, c, /*reuse_a=*/false, /*reuse_b=*/false);
}

// Kernel 1: per-(batch,chunk) partial Gram matrices of W = [V | E | 0pad].
__global__ __launch_bounds__(256)
void gram_partial_kernel(const float* __restrict__ emb,   // (B, N, 40)
                         const float* __restrict__ asn,   // (B, N, 4)
                         float* __restrict__ partial)     // (B*NCHUNK, 1536)
{
    const int tid  = threadIdx.x;
    const int lane = tid & 31;
    const int wv   = tid >> 5;
    const int b     = blockIdx.x / NCHUNK;
    const int chunk = blockIdx.x % NCHUNK;

    const int m    = lane & 15;        // M (A) == N (B) index within tile
    const int kb   = (lane >> 4) << 1; // K base: 0 for lanes 0-15, 2 for 16-31

    const int row0 = chunk * ROWS_PER_BLOCK + wv * ROWS_PER_WAVE;

    const float* ebase = emb + (size_t)b * NN * DD;
    const float* abase = asn + (size_t)b * NN * SS;

    // Per-lane, per-tile load pointers for W[row][c], c = 16*tile + m.
    // c < 4 -> assignments; 4 <= c < 44 -> embeddings; c >= 44 -> 0 (clamp+scale).
    const float* p0[3];
    const float* p1[3];
    int stepf[3];
    float sc2 = 1.0f;  // only tile 2 can hit the pad region
#pragma unroll
    for (int i = 0; i < 3; ++i) {
        const int c = 16 * i + m;
        const float* base;
        int st;
        if (c < SS) {
            base = abase + c;
            st = SS;
        } else {
            int ce = c - SS;
            if (ce >= DD) { ce = DD - 1; if (i == 2) sc2 = 0.0f; }
            base = ebase + ce;
            st = DD;
        }
        p0[i] = base + (size_t)(row0 + kb) * st;  // row kb   -> frag VGPR 0
        p1[i] = p0[i] + st;                       // row kb+1 -> frag VGPR 1
        stepf[i] = 4 * st;                        // advance 4 rows per step
    }
    const float* q00 = p0[0]; const float* q01 = p1[0];
    const float* q10 = p0[1]; const float* q11 = p1[1];
    const float* q20 = p0[2]; const float* q21 = p1[2];
    const int s0 = stepf[0], s1 = stepf[1], s2 = stepf[2];

    v8f c00 = {}, c01 = {}, c02 = {}, c11 = {}, c12 = {}, c22 = {};

#pragma unroll 4
    for (int s = 0; s < NSTEPS; ++s) {
        v2f f0, f1, f2;
        f0.x = q00[0];        f0.y = q01[0];
        f1.x = q10[0];        f1.y = q11[0];
        f2.x = q20[0] * sc2;  f2.y = q21[0] * sc2;

        // Gram symmetry: A-frag and B-frag register images coincide.
        c00 = wmma_f32(f0, f0, c00);
        c01 = wmma_f32(f0, f1, c01);
        c02 = wmma_f32(f0, f2, c02);
        c11 = wmma_f32(f1, f1, c11);
        c12 = wmma_f32(f1, f2, c12);
        c22 = wmma_f32(f2, f2, c22);

        q00 += s0; q01 += s0;
        q10 += s1; q11 += s1;
        q20 += s2; q21 += s2;
    }

    // Deterministic cross-wave reduction: each wave dumps its 6 tiles to LDS,
    // then thread tid sums position tid of each tile across the 8 waves.
    __shared__ float red[NTILE][NWAVES][TILEVALS];   // 48 KB
    v8f accs[NTILE] = { c00, c01, c02, c11, c12, c22 };
#pragma unroll
    for (int t = 0; t < NTILE; ++t)
#pragma unroll
        for (int k = 0; k < 8; ++k)
            red[t][wv][k * 32 + lane] = accs[t][k];
    __syncthreads();

    float* out = partial + (size_t)blockIdx.x * PARTIAL_FLOATS;
#pragma unroll
    for (int t = 0; t < NTILE; ++t) {
        float ssum = 0.0f;
#pragma unroll
        for (int ww = 0; ww < NWAVES; ++ww)
            ssum += red[t][ww][tid];
        out[t * TILEVALS + tid] = ssum;
    }
}

// Kernel 2: sum partials per batch, apply block signs, square, reduce to scalar.
__global__ __launch_bounds__(256)
void gram_reduce_kernel(const float* __restrict__ partial, float* __restrict__ out)
{
    const int tid  = threadIdx.x;
    const int k    = tid >> 5;          // C-matrix VGPR index 0..7
    const int lane = tid & 31;
    const int half = lane >> 4;
    const int m    = lane & 15;

    const int ti_tab[NTILE] = {0, 0, 0, 1, 1, 2};
    const int tj_tab[NTILE] = {0, 1, 2, 1, 2, 2};

    float acc = 0.0f;
#pragma unroll
    for (int t = 0; t < NTILE; ++t) {
        const int r = 16 * ti_tab[t] + k + 8 * half;   // C layout: M = k (+8 hi half)
        const int c = 16 * tj_tab[t] + m;
        const float w    = ((r < SS) == (c < SS)) ? 1.0f : -1.0f;  // mixed V/E -> -1
        const float mult = (ti_tab[t] == tj_tab[t]) ? 1.0f : 2.0f; // off-diag tile x2
        const float wm = w * mult;
        for (int b = 0; b < BB; ++b) {
            float g = 0.0f;
            for (int ch = 0; ch < NCHUNK; ++ch)
                g += partial[(size_t)(b * NCHUNK + ch) * PARTIAL_FLOATS
                             + t * TILEVALS + tid];
            acc += wm * g * g;
        }
    }

    __shared__ float red[256];
    red[tid] = acc;
    __syncthreads();
    for (int s = 128; s > 0; s >>= 1) {
        if (tid < s) red[tid] += red[tid + s];
        __syncthreads();
    }
    if (tid == 0)
        out[0] = red[0] * (1.0f / (float)((size_t)BB * NN));
}

extern "C" void kernel_launch(void* const* d_in, const int* in_sizes, int n_in,
                              void* d_out, int out_size, void* d_ws, size_t ws_size,
                              hipStream_t stream) {
    const float* emb = (const float*)d_in[0];  // embeddings (B,F,T,D) f32
    const float* asn = (const float*)d_in[1];  // assignments (B,F,T,S) f32
    float* partial = (float*)d_ws;             // needs BB*NCHUNK*1536*4 = 3 MB

    gram_partial_kernel<<<BB * NCHUNK, 256, 0, stream>>>(emb, asn, partial);
    gram_reduce_kernel<<<1, 256, 0, stream>>>(partial, (float*)d_out);
}